// HistLayer_7430293422087
// MI455X (gfx1250) — compile-verified
//
#include <hip/hip_runtime.h>
#include <cstdint>

// ---------------------------------------------------------------------------
// HistLayer on MI455X (gfx1250).
// Roofline: 8*256*256*64*4B = 134MB read, 16KB written -> pure HBM-bound,
// ~5.7us at 23.3 TB/s. Strategy:
//   * stream x via async global->LDS (ASYNCcnt) double buffering (CDNA5 path),
//     compact software-pipelined loop (no full unroll -> fits I$)
//   * per element: 1 bin via fma+cvt (bins partition the range), y=1.01^z via
//     2nd-order Taylor (|err|<1e-10), branchless select, ds_add_f32 into
//     bank-padded replicated LDS histograms
//   * per-block finalize: replica (K=4) reduction via V_WMMA_F32_16X16X4_F32
//     with B=ones (layout-independent), then native global f32 atomics
// ---------------------------------------------------------------------------

#define NB       8
#define CCH      64
#define HWTOT    65536
#define SPB      1024              // spatial positions per block
#define ITERS    64                // SPB / 16 spatial per block-iteration
#define HSTRIDE  9                 // padded per-channel stride (9 coprime 64)
#define RSTRIDE  (CCH * HSTRIDE)   // 576 floats per replica
#define NREP     4
#define GSTEP    (16 * CCH * 4)    // bytes advanced per iteration (4096)

// bin constants: edges = linspace(-0.05, 1.05, 9), step = 0.1375
#define INV_STEP 7.2727272727f     // 1/0.1375
#define T_OFF    0.36363636f       // 0.05/0.1375
#define STEP     0.1375f
#define C0       0.01875f          // first center
#define HALF_W   0.06875f
#define LN101    0.00995033085f    // ln(1.01)
#define SCALE    (1.0f / 65536.0f)

typedef __attribute__((ext_vector_type(2))) float v2f;
typedef __attribute__((ext_vector_type(8))) float v8f;

#define ASYNC_LOAD_B128(ldsoff, gaddr)                                       \
  asm volatile("global_load_async_to_lds_b128 %0, %1, off"                   \
               :: "v"(ldsoff), "v"(gaddr) : "memory")
#define WAIT_ASYNC_1() asm volatile("s_wait_asynccnt 0x1" ::: "memory")
#define WAIT_ASYNC_0() asm volatile("s_wait_asynccnt 0x0" ::: "memory")

__global__ void zero_out_kernel(float* __restrict__ o, int n) {
  int i = blockIdx.x * blockDim.x + threadIdx.x;
  if (i < n) o[i] = 0.0f;
}

// Branchless per-float4 histogram update: ~12 VALU + 1 ds_add per element.
__device__ __forceinline__ void accum4(float4 v, int hBase,
                                       float* __restrict__ hist) {
#pragma unroll
  for (int j = 0; j < 4; ++j) {
    float xv = (&v.x)[j];
    float t  = __builtin_fmaf(xv, INV_STEP, T_OFF);   // (x+0.05)/step
    int   b  = (int)t;                                // t in (0,8): trunc==floor
    float bf = (float)b;
    float ce = __builtin_fmaf(bf, STEP, C0);          // bin center
    float z  = HALF_W - fabsf(xv - ce);
    float w  = z * LN101;
    float y  = __builtin_fmaf(w, __builtin_fmaf(w, 0.5f, 1.0f), 1.0f);
    float yc = (z > 0.0f) ? y : 0.0f;                 // ThresholdedReLU(1)
    atomicAdd(&hist[hBase + j * HSTRIDE + b], yc);    // ds_add_f32, EXEC full
  }
}

__global__ __launch_bounds__(256) void hist_kernel(
    const float* __restrict__ x, float* __restrict__ out) {
  __shared__ float4 staging[2][8][32];            // 8 KB, per-wave dbl buffer
  __shared__ float  hist[NREP * RSTRIDE];         // 9216 B, 4 replicas

  const int tid   = threadIdx.x;
  const int w     = tid >> 5;                     // wave in block (0..7)
  const int l     = tid & 31;                     // lane
  const int sHalf = l >> 4;                       // spatial within wave pair
  const int c4    = (l & 15) * 4;                 // channel base (float4)
  const int n     = blockIdx.y;
  const int rep   = w & 3;
  const int hBase = rep * RSTRIDE + c4 * HSTRIDE;

  for (int i = tid; i < NREP * RSTRIDE; i += 256) hist[i] = 0.0f;
  __syncthreads();

  // lane's first element: x[n][s0][c4..c4+3], NHWC flat
  const int s0 = blockIdx.x * SPB + w * 2 + sHalf;
  const float* p0 = x + ((size_t)n << 22) + (size_t)s0 * CCH + c4;
  const uint64_t ga0  = (uint64_t)(uintptr_t)p0;
  const uint32_t lds0 = (uint32_t)(uintptr_t)&staging[0][w][l];
  const uint32_t lds1 = (uint32_t)(uintptr_t)&staging[1][w][l];

  // Software pipeline, 2 outstanding async loads, compact loop body.
  // Invariant at loop top: loads k (buf0) and k+1 (buf1) are in flight.
  ASYNC_LOAD_B128(lds0, ga0);
  ASYNC_LOAD_B128(lds1, ga0 + GSTEP);
#pragma unroll 1
  for (int k = 0; k < ITERS - 2; k += 2) {
    uint64_t gnext = ga0 + (uint64_t)(k + 2) * GSTEP;
    WAIT_ASYNC_1();                               // load k complete (in-order)
    accum4(staging[0][w][l], hBase, hist);        // consume buf0 fully...
    ASYNC_LOAD_B128(lds0, gnext);                 // ...then recycle it
    WAIT_ASYNC_1();                               // load k+1 complete
    accum4(staging[1][w][l], hBase, hist);
    ASYNC_LOAD_B128(lds1, gnext + GSTEP);
  }
  WAIT_ASYNC_1();
  accum4(staging[0][w][l], hBase, hist);          // iter ITERS-2
  WAIT_ASYNC_0();
  accum4(staging[1][w][l], hBase, hist);          // iter ITERS-1

  __syncthreads();

  // ---- finalize: sum 4 replicas, scale by 1/HW, atomically add to out ----
#if defined(__gfx1250__) && __has_builtin(__builtin_amdgcn_wmma_f32_16x16x4_f32)
  {
    // K=4 replica reduction as D = A(16x4) * ones(4x16).
    // A layout (ISA 7.12.2): lanes 0-15 hold K=0(v0),K=1(v1); lanes 16-31
    // hold K=2(v0),K=3(v1); M = lane&15. B = ones => D[m][*] = sum_k A[m][k].
    const int m = l & 15, hh = l >> 4;
#pragma unroll
    for (int i = 0; i < 4; ++i) {
      int e    = w * 64 + i * 16 + m;                 // output elem = c*8+b
      int hoff = (e >> 3) * HSTRIDE + (e & 7);
      v2f a;  a.x = hist[(2 * hh + 0) * RSTRIDE + hoff];
              a.y = hist[(2 * hh + 1) * RSTRIDE + hoff];
      v2f bo; bo.x = 1.0f; bo.y = 1.0f;
      v8f c = {};
      c = __builtin_amdgcn_wmma_f32_16x16x4_f32(
          false, a, false, bo, (short)0, c, false, false);
      // D layout: vgpr r: lanes 0-15 -> M=r, lanes 16-31 -> M=r+8 (any N)
      if (m == 0) {
        int eb = w * 64 + i * 16 + hh * 8;
#pragma unroll
        for (int r = 0; r < 8; ++r)
          atomicAdd(&out[n * 512 + eb + r], c[r] * SCALE);
      }
    }
  }
#else
  for (int e = tid; e < 512; e += 256) {
    int hoff = (e >> 3) * HSTRIDE + (e & 7);
    float s = hist[hoff] + hist[RSTRIDE + hoff] +
              hist[2 * RSTRIDE + hoff] + hist[3 * RSTRIDE + hoff];
    atomicAdd(&out[n * 512 + e], s * SCALE);
  }
#endif
}

extern "C" void kernel_launch(void* const* d_in, const int* in_sizes, int n_in,
                              void* d_out, int out_size, void* d_ws,
                              size_t ws_size, hipStream_t stream) {
  (void)in_sizes; (void)n_in; (void)d_ws; (void)ws_size;
  const float* x = (const float*)d_in[0];   // [8,256,256,64] f32
  float* out = (float*)d_out;               // [8,512] f32

  zero_out_kernel<<<dim3((out_size + 255) / 256), 256, 0, stream>>>(out,
                                                                    out_size);
  hist_kernel<<<dim3(HWTOT / SPB, 8), 256, 0, stream>>>(x, out);
}